// ExampleModel_35596688949292
// MI455X (gfx1250) — compile-verified
//
#include <hip/hip_runtime.h>
#include <stdint.h>

// MoE gating: softmax(131072x256) -> top-8 weights + indices per token.
// Memory-bound (134 MB in / 8.4 MB out ~= 6.1 us at 23.3 TB/s). CDNA5 path:
// async global->LDS double buffering (GLOBAL_LOAD_ASYNC_TO_LDS_B128 +
// s_wait_asynccnt) streams rows while wave32 does softmax + iterative
// argmax in registers. Results are latched per-lane and written with two
// coalesced stores per token (instead of 16 divergent single-lane stores).

#define EXPERTS 256
#define TOPK 8
#define WAVES_PER_BLOCK 8
#define BLOCK_THREADS (WAVES_PER_BLOCK * 32)

// Issue one token row (256 floats = 1024B) into LDS asynchronously.
// Each lane moves 16B from the low half and 16B from the high half; the
// instruction IOFFSET is added to BOTH the global and LDS addresses
// (CDNA5 ISA), so offset:512 reuses the same base regs.
__device__ __forceinline__ void async_row_to_lds(const float* gsrc,
                                                 unsigned lds_byte_addr) {
  unsigned long long ga = (unsigned long long)(uintptr_t)gsrc;
  asm volatile(
      "global_load_async_to_lds_b128 %0, %1, off\n\t"
      "global_load_async_to_lds_b128 %0, %1, off offset:512"
      :
      : "v"(lds_byte_addr), "v"(ga)
      : "memory");
}

__global__ __launch_bounds__(BLOCK_THREADS) void moe_topk_softmax_kernel(
    const float* __restrict__ gating, float* __restrict__ out_w,
    float* __restrict__ out_i, int num_tokens) {
  // per-wave double buffer: 8 waves * 2 * 1KB = 16 KB LDS
  __shared__ __align__(16) float rows[WAVES_PER_BLOCK][2][EXPERTS];

  const int lane = threadIdx.x & 31;
  const int wave = threadIdx.x >> 5;
  const int wid = blockIdx.x * WAVES_PER_BLOCK + wave;
  const int nwaves = gridDim.x * WAVES_PER_BLOCK;

  int buf = 0;
  if (wid < num_tokens) {
    async_row_to_lds(gating + (size_t)wid * EXPERTS + (lane << 2),
                     (unsigned)(uintptr_t)&rows[wave][0][lane << 2]);
  }

  for (int t = wid; t < num_tokens; t += nwaves) {
    const int nxt = t + nwaves;
    float* ldsbase = &rows[wave][buf][0];
    if (nxt < num_tokens) {
      // prefetch next token into the other buffer, then wait only for the
      // current buffer's two async ops (in-order completion -> cnt<=2).
      async_row_to_lds(gating + (size_t)nxt * EXPERTS + (lane << 2),
                       (unsigned)(uintptr_t)&rows[wave][buf ^ 1][lane << 2]);
      asm volatile("s_wait_asynccnt 0x2" ::"v"(ldsbase) : "memory");
    } else {
      asm volatile("s_wait_asynccnt 0x0" ::"v"(ldsbase) : "memory");
    }

    // Lane owns experts {4*lane..4*lane+3} and {128+4*lane..128+4*lane+3}
    // (matches the conflict-free lane*16B async layout).
    const float4* lp = (const float4*)&rows[wave][buf][lane << 2];
    const float4 a = lp[0];
    const float4 b = lp[32];  // +128 floats
    float v[8] = {a.x, a.y, a.z, a.w, b.x, b.y, b.z, b.w};
    int ei[8];
#pragma unroll
    for (int j = 0; j < 4; ++j) {
      ei[j] = (lane << 2) + j;
      ei[j + 4] = 128 + (lane << 2) + j;
    }

    // ---- softmax statistics: wave32 butterfly reductions (5 steps) ----
    float m = v[0];
#pragma unroll
    for (int j = 1; j < 8; ++j) m = fmaxf(m, v[j]);
#pragma unroll
    for (int off = 16; off >= 1; off >>= 1)
      m = fmaxf(m, __shfl_xor(m, off, 32));

    float s = 0.0f;
#pragma unroll
    for (int j = 0; j < 8; ++j) s += __expf(v[j] - m);
#pragma unroll
    for (int off = 16; off >= 1; off >>= 1) s += __shfl_xor(s, off, 32);
    const float invs = 1.0f / s;

    // ---- top-8 by iterative wave argmax over raw logits ----
    // softmax is monotone, so selecting on logits == selecting on softmax;
    // tie-break: smaller expert index wins (matches jax.lax.top_k order).
    // Lane k latches round-k's result; stores are coalesced after the loop.
    float my_w = 0.0f;
    float my_i = 0.0f;
#pragma unroll
    for (int k = 0; k < TOPK; ++k) {
      float bv = v[0];
      int bi = ei[0];
#pragma unroll
      for (int j = 1; j < 8; ++j) {
        if (v[j] > bv) {  // strict '>' keeps the smallest index in-lane
          bv = v[j];
          bi = ei[j];
        }
      }
#pragma unroll
      for (int off = 16; off >= 1; off >>= 1) {
        const float ov = __shfl_xor(bv, off, 32);
        const int oi = __shfl_xor(bi, off, 32);
        if (ov > bv || (ov == bv && oi < bi)) {
          bv = ov;
          bi = oi;
        }
      }
      // branch-free latch: only lane k keeps round k's result
      if (lane == k) {
        my_w = __expf(bv - m) * invs;
        my_i = (float)bi;
      }
      // remove winner from its owning lane's register file (no dyn index)
#pragma unroll
      for (int j = 0; j < 8; ++j)
        if (ei[j] == bi) v[j] = -__builtin_inff();
    }

    // two coalesced stores per token (lanes 0..7 active)
    if (lane < TOPK) {
      const size_t o = (size_t)t * TOPK + lane;
      out_w[o] = my_w;
      out_i[o] = my_i;  // index output into the flat float buffer
    }

    buf ^= 1;
  }
}

extern "C" void kernel_launch(void* const* d_in, const int* in_sizes, int n_in,
                              void* d_out, int out_size, void* d_ws,
                              size_t ws_size, hipStream_t stream) {
  (void)n_in;
  (void)d_ws;
  (void)ws_size;
  (void)out_size;
  const float* gating = (const float*)d_in[0];
  const int num_tokens = in_sizes[0] / EXPERTS;  // 131072
  float* out_w = (float*)d_out;
  float* out_i = out_w + (size_t)num_tokens * TOPK;

  // ~8 tokens per wave so the async double-buffer pipeline has depth.
  int blocks = (num_tokens + WAVES_PER_BLOCK * 8 - 1) / (WAVES_PER_BLOCK * 8);
  if (blocks > 2048) blocks = 2048;
  if (blocks < 1) blocks = 1;

  moe_topk_softmax_kernel<<<dim3(blocks), dim3(BLOCK_THREADS), 0, stream>>>(
      gating, out_w, out_i, num_tokens);
}